// SoftPool2d_50551765074342
// MI455X (gfx1250) — compile-verified
//
#include <hip/hip_runtime.h>

// SoftPool2d: x (16,96,224,224) f32, k=3x3, s=2, p=1 -> (16,96,112,112)
// Memory-bound: ~385 MB total traffic -> ~16.5 us floor @ 23.3 TB/s.
// Strategy: async global->LDS staging (CDNA5 ASYNCcnt path), native v_exp_f32.

#define N_    16
#define C_    96
#define H_    224
#define W_    224
#define HO_   112
#define WO_   112
#define TH_   16                 // output rows per block
#define ROWS_ (2 * TH_ + 1)      // 33 input rows per tile
#define PITCH_ 232               // floats per LDS row (16B-aligned row starts)
#define SEGS_ (W_ / 4)           // 56 float4 segments per row
#define NSEG_ (ROWS_ * SEGS_)    // 1848 transfers per block

#if __has_builtin(__builtin_amdgcn_global_load_async_to_lds_b128)
#define HAVE_ASYNC_LDS 1
#else
#define HAVE_ASYNC_LDS 0
#endif

typedef float float4v __attribute__((ext_vector_type(4)));
typedef int   v4i_t   __attribute__((ext_vector_type(4)));
// Pointee types carrying explicit address spaces (clang prints AS1 as
// "__device__", AS3 as "__shared__" in HIP diagnostics).
typedef __attribute__((address_space(1))) v4i_t gv4i;   // global int4
typedef __attribute__((address_space(3))) v4i_t lv4i;   // LDS int4

__device__ __forceinline__ void copy16_g2l(const float* __restrict__ g,
                                           float* __restrict__ l) {
#if HAVE_ASYNC_LDS
  // global_load_async_to_lds_b128: per-lane 16B DMA, tracked by ASYNCcnt.
  __builtin_amdgcn_global_load_async_to_lds_b128(
      (gv4i*)(g), (lv4i*)(l), /*imm offset*/ 0, /*cpol*/ 0);
#else
  *(float4v*)l = *(const float4v*)g;
#endif
}

__device__ __forceinline__ void wait_async_copies() {
#if HAVE_ASYNC_LDS
#if __has_builtin(__builtin_amdgcn_s_wait_asynccnt)
  __builtin_amdgcn_s_wait_asynccnt(0);
#else
  asm volatile("s_wait_asynccnt 0" ::: "memory");
#endif
#endif
}

__device__ __forceinline__ float fast_exp2(float x) {
#if __has_builtin(__builtin_amdgcn_exp2f)
  return __builtin_amdgcn_exp2f(x);   // v_exp_f32
#else
  return exp2f(x);
#endif
}

__device__ __forceinline__ float fast_rcp(float x) {
#if __has_builtin(__builtin_amdgcn_rcpf)
  return __builtin_amdgcn_rcpf(x);    // v_rcp_f32
#else
  return 1.0f / x;
#endif
}

__global__ __launch_bounds__(256) void softpool2d_kernel(
    const float* __restrict__ x, const float* __restrict__ raw_alpha,
    float* __restrict__ out) {
  __shared__ float tile[ROWS_ * PITCH_];

  const int nc  = blockIdx.x;       // 0..1535  (n*C + c)
  const int ty  = blockIdx.y;       // 0..6
  const int oh0 = ty * TH_;
  const int ih0 = 2 * oh0 - 1;      // first input row of tile (may be -1)
  const int t   = threadIdx.x;

  const float* plane = x + (size_t)nc * (size_t)(H_ * W_);

  // Zero only what is actually read: left-pad column (lds col 3) on every
  // row, and the whole top row when ih0 < 0 (only tile ty==0).
  if (t < ROWS_) tile[t * PITCH_ + 3] = 0.0f;
  if (ih0 < 0) {
    for (int c = t; c < PITCH_; c += 256) tile[c] = 0.0f;
  }
  __syncthreads();

  // Stage 33 rows x 224 floats into LDS cols [4, 228) via async 16B copies.
  #pragma unroll
  for (int k = 0; k < (NSEG_ + 255) / 256; ++k) {
    int i = t + k * 256;
    if (i < NSEG_) {
      int r  = i / SEGS_;
      int s  = i - r * SEGS_;
      int ih = ih0 + r;
      if (ih >= 0) {  // only false for r==0 in the ty==0 tile
        copy16_g2l(plane + (size_t)ih * W_ + s * 4,
                   &tile[r * PITCH_ + 4 + s * 4]);
      }
    }
  }
  wait_async_copies();   // my waves' DMAs done
  __syncthreads();       // everyone's DMAs visible

  // alpha = softplus(raw_alpha); scale folds ln->log2 for native exp2.
  const float ra    = raw_alpha[0];
  const float alpha = __logf(1.0f + __expf(ra));
  const float sc    = alpha * 1.44269504088896340736f;  // alpha * log2(e)

  float* orow = out + (size_t)nc * (size_t)(HO_ * WO_) + (size_t)oh0 * WO_;

  // 16*112 = 1792 outputs per block -> 7 per thread.
  #pragma unroll
  for (int k = 0; k < (TH_ * WO_) / 256; ++k) {
    int idx = t + k * 256;
    int r   = idx / WO_;            // local output row 0..15
    int ow  = idx - r * WO_;        // 0..111
    const float* base = &tile[(2 * r) * PITCH_ + 2 * ow + 3];
    float num = 0.0f, den = 0.0f;
    #pragma unroll
    for (int ky = 0; ky < 3; ++ky) {
      #pragma unroll
      for (int kx = 0; kx < 3; ++kx) {
        float v = base[ky * PITCH_ + kx];
        float e = fast_exp2(sc * v);
        den += e;
        num = fmaf(v, e, num);
      }
    }
    float res = num * fast_rcp(den);
    __builtin_nontemporal_store(res, &orow[(size_t)r * WO_ + ow]);
  }
}

extern "C" void kernel_launch(void* const* d_in, const int* in_sizes, int n_in,
                              void* d_out, int out_size, void* d_ws, size_t ws_size,
                              hipStream_t stream) {
  (void)in_sizes; (void)n_in; (void)out_size; (void)d_ws; (void)ws_size;
  const float* x  = (const float*)d_in[0];
  const float* ra = (const float*)d_in[1];
  float* out      = (float*)d_out;

  dim3 grid(N_ * C_, HO_ / TH_, 1);   // (1536, 7)
  dim3 block(256, 1, 1);              // 8 waves (wave32)
  softpool2d_kernel<<<grid, block, 0, stream>>>(x, ra, out);
}